// Attention_4904852652609
// MI455X (gfx1250) — compile-verified
//
#include <hip/hip_runtime.h>
#include <math.h>

// Problem constants (reference: B=16, S=4096, E=1024, D=1024, WINDOW_WIDTH=2.0)
#define BB 16
#define SS 4096
#define EE 1024
#define DD 1024
#define CS 128               // s-rows per fused chunk
#define NCH (SS / CS)        // 32 chunks per batch
#define SUB 16               // rows per online-softmax sub-tile

typedef float v2f __attribute__((ext_vector_type(2)));
typedef float v8f __attribute__((ext_vector_type(8)));

// ---------------------------------------------------------------------------
// u = dec @ Wa_w   (u[b,e] = sum_d dec[b,d] * Wa_w[d,e])  via V_WMMA_F32_16X16X4_F32
// fp32 A 16x4 layout: lanes 0-15 -> M, VGPR{0,1} = K{0,1}; lanes 16-31 -> K{2,3}.
// ---------------------------------------------------------------------------
__global__ void wmma_u_gemm(const float* __restrict__ dec,   // [B,D]
                            const float* __restrict__ Wa,    // [D,E] row-major
                            float* __restrict__ u)           // [B,E]
{
    const int n0   = blockIdx.x * 16;
    const int lane = threadIdx.x;      // 0..31
    const int hi   = lane >> 4;
    const int l16  = lane & 15;
    v8f acc = {};
    for (int k = 0; k < DD; k += 4) {
        const int ka = k + hi * 2;
        v2f a, b;
        a.x = dec[l16 * DD + ka];
        a.y = dec[l16 * DD + ka + 1];
        b.x = Wa[(size_t)(ka)     * EE + n0 + l16];
        b.y = Wa[(size_t)(ka + 1) * EE + n0 + l16];
        acc = __builtin_amdgcn_wmma_f32_16x16x4_f32(
            false, a, false, b, (short)0, acc, false, false);
    }
    #pragma unroll
    for (int r = 0; r < 8; ++r)
        u[(hi * 8 + r) * EE + n0 + l16] = acc[r];
}

// ---------------------------------------------------------------------------
// h = tanh(dec @ Wp_w.T + Wp_b)
// ---------------------------------------------------------------------------
__global__ void wmma_h_gemm(const float* __restrict__ dec,   // [B,D]
                            const float* __restrict__ Wp,    // [D,D] row-major
                            const float* __restrict__ Wpb,   // [D]
                            float* __restrict__ h)           // [B,D]
{
    const int n0   = blockIdx.x * 16;
    const int lane = threadIdx.x;
    const int hi   = lane >> 4;
    const int l16  = lane & 15;
    v8f acc = {};
    for (int k = 0; k < DD; k += 4) {
        const int ka = k + hi * 2;
        v2f a, b;
        a.x = dec[l16 * DD + ka];
        a.y = dec[l16 * DD + ka + 1];
        b.x = Wp[(size_t)(n0 + l16) * DD + ka];       // B[k,n] = Wp[n,k]
        b.y = Wp[(size_t)(n0 + l16) * DD + ka + 1];
        acc = __builtin_amdgcn_wmma_f32_16x16x4_f32(
            false, a, false, b, (short)0, acc, false, false);
    }
    const float bias = Wpb[n0 + l16];
    #pragma unroll
    for (int r = 0; r < 8; ++r)
        h[(hi * 8 + r) * DD + n0 + l16] = tanhf(acc[r] + bias);
}

// ---------------------------------------------------------------------------
// p[b] = S * sigmoid(h[b,:]·vp_w + vp_b);  c[b] = Wa_b·dec[b,:]
// ---------------------------------------------------------------------------
__global__ void pos_c_kernel(const float* __restrict__ h,
                             const float* __restrict__ vpw,
                             const float* __restrict__ vpb,
                             const float* __restrict__ Wab,
                             const float* __restrict__ dec,
                             float* __restrict__ p, float* __restrict__ c)
{
    const int b    = threadIdx.x >> 5;   // 16 waves -> 16 batches
    const int lane = threadIdx.x & 31;
    float s1 = 0.f, s2 = 0.f;
    for (int j = lane; j < DD; j += 32) {
        s1 += h[b * DD + j] * vpw[j];
        s2 += Wab[j] * dec[b * DD + j];
    }
    #pragma unroll
    for (int off = 16; off; off >>= 1) {
        s1 += __shfl_xor(s1, off, 32);
        s2 += __shfl_xor(s2, off, 32);
    }
    if (lane == 0) {
        const float t = s1 + vpb[0];
        p[b] = (float)SS / (1.0f + __expf(-t));
        c[b] = s2;
    }
}

// ---------------------------------------------------------------------------
// Fused att + online-softmax + windowed awe accumulation (single HBM pass).
// Block = (b, 128-row chunk); processed in 16-row sub-tiles:
//   phase A: att[s] = enc[b,s,:]·u[b] + c[b]     (wave-per-row, HBM read)
//   phase B: running (m, Z) update + rescale; acc[e] += e^{att-m}·gauss·enc
//            (re-reads the 16 rows just touched -> L1/WGP$ hit)
// Emits per-chunk (m, Z, acc[1024]) and raw att (for the alpha output).
// ---------------------------------------------------------------------------
__global__ void fused_att_awe(const float* __restrict__ enc,
                              const float* __restrict__ u,
                              const float* __restrict__ c,
                              const float* __restrict__ p,
                              float* __restrict__ att_out,   // [B,S] (alpha region)
                              float* __restrict__ m_out,     // [B*NCH]
                              float* __restrict__ z_out,     // [B*NCH]
                              float* __restrict__ A_out)     // [B*NCH, E]
{
    const int b    = blockIdx.x >> 5;          // / NCH
    const int ch   = blockIdx.x & (NCH - 1);
    const int s0   = ch * CS;
    const int t    = threadIdx.x;              // 256 threads = 8 waves
    const int wave = t >> 5;
    const int lane = t & 31;
    const int e0   = t * 4;                    // column slice for accumulation

    __shared__ float att_s[SUB];
    __shared__ float w_s[SUB];                 // e^{att-m} * gauss
    __shared__ float e_s[SUB];                 // e^{att-m}

    const float cb = c[b];
    const float pb = p[b];
    const float* encb = enc + (size_t)b * SS * EE;

    // preload u[b,:] slice for the dot products (8 x float4 per lane)
    const float4* u4 = (const float4*)(u + b * EE);
    float4 ur[8];
    #pragma unroll
    for (int i = 0; i < 8; ++i) ur[i] = u4[lane + 32 * i];

    float m = -INFINITY, Z = 0.f;
    float4 acc = {0.f, 0.f, 0.f, 0.f};

    for (int sub = 0; sub < CS / SUB; ++sub) {
        const int sbase = s0 + sub * SUB;

        // ---- phase A: 16 rows of att, wave w handles rows 2w, 2w+1 ----
        #pragma unroll
        for (int j = 0; j < 2; ++j) {
            const int r = wave * 2 + j;
            const int s = sbase + r;
            const float4* e4 = (const float4*)(encb + (size_t)s * EE);
            float d = 0.f;
            #pragma unroll
            for (int i = 0; i < 8; ++i) {
                const float4 ev = e4[lane + 32 * i];
                d += ev.x * ur[i].x + ev.y * ur[i].y + ev.z * ur[i].z + ev.w * ur[i].w;
            }
            #pragma unroll
            for (int off = 16; off; off >>= 1) d += __shfl_xor(d, off, 32);
            if (lane == 0) {
                const float a = d + cb;
                att_s[r] = a;
                att_out[b * SS + s] = a;       // raw att, finalized to alpha later
            }
        }
        __syncthreads();

        // ---- online softmax update (uniform across block) ----
        float mx = att_s[0];
        #pragma unroll
        for (int r = 1; r < SUB; ++r) mx = fmaxf(mx, att_s[r]);
        const float m_new = fmaxf(m, mx);
        const float scale = __expf(m - m_new);  // 0 on first iter (m = -inf)

        if (t < SUB) {
            const float a  = att_s[t];
            const float ez = __expf(a - m_new);
            const float dd = (float)(sbase + t) - pb;
            e_s[t] = ez;
            w_s[t] = ez * __expf(-(dd * dd) * 0.125f);   // gauss, W=2
        }
        __syncthreads();

        float zs = 0.f;
        #pragma unroll
        for (int r = 0; r < SUB; ++r) zs += e_s[r];
        Z = Z * scale + zs;
        m = m_new;

        // ---- phase B: weighted accumulation (rows are cache-hot) ----
        acc.x *= scale; acc.y *= scale; acc.z *= scale; acc.w *= scale;
        const float* rowbase = encb + (size_t)sbase * EE + e0;
        #pragma unroll
        for (int r = 0; r < SUB; ++r) {
            const float wr = w_s[r];
            const float4 ev = *(const float4*)(rowbase + (size_t)r * EE);
            acc.x += wr * ev.x; acc.y += wr * ev.y;
            acc.z += wr * ev.z; acc.w += wr * ev.w;
        }
        __syncthreads();   // att_s/e_s/w_s reused next sub-tile
    }

    *(float4*)(A_out + (size_t)blockIdx.x * EE + e0) = acc;
    if (t == 0) { m_out[blockIdx.x] = m; z_out[blockIdx.x] = Z; }
}

// ---------------------------------------------------------------------------
// Per-batch combine of chunk partials: M, Z, awe; saves (M,Z) for alpha pass.
// ---------------------------------------------------------------------------
__global__ void combine_kernel(const float* __restrict__ A,
                               const float* __restrict__ marr,
                               const float* __restrict__ zarr,
                               float* __restrict__ awe,
                               float* __restrict__ MZ)
{
    const int b = blockIdx.x;
    const int t = threadIdx.x;           // 256
    __shared__ float sm[NCH], sz[NCH];
    if (t < NCH) { sm[t] = marr[b * NCH + t]; sz[t] = zarr[b * NCH + t]; }
    __syncthreads();

    float M = sm[0];
    #pragma unroll
    for (int ch = 1; ch < NCH; ++ch) M = fmaxf(M, sm[ch]);
    float Z = 0.f;
    #pragma unroll
    for (int ch = 0; ch < NCH; ++ch) Z += sz[ch] * __expf(sm[ch] - M);

    const float4* A4 = (const float4*)A;
    float4 acc = {0.f, 0.f, 0.f, 0.f};
    #pragma unroll
    for (int ch = 0; ch < NCH; ++ch) {
        const float scl = __expf(sm[ch] - M);
        const float4 ev = A4[(size_t)(b * NCH + ch) * (EE / 4) + t];
        acc.x += scl * ev.x; acc.y += scl * ev.y;
        acc.z += scl * ev.z; acc.w += scl * ev.w;
    }
    const float invZ = 1.0f / Z;
    acc.x *= invZ; acc.y *= invZ; acc.z *= invZ; acc.w *= invZ;
    ((float4*)awe)[b * (EE / 4) + t] = acc;

    if (t == 0) { MZ[2 * b] = M; MZ[2 * b + 1] = Z; }
}

// ---------------------------------------------------------------------------
// alpha[b,s] = e^{att-M}/Z * gauss   (in-place on staged att, 256 KB)
// ---------------------------------------------------------------------------
__global__ void alpha_kernel(float* __restrict__ alpha,
                             const float* __restrict__ MZ,
                             const float* __restrict__ p)
{
    const int b = blockIdx.x;
    const int t = threadIdx.x;
    const float M    = MZ[2 * b];
    const float invZ = 1.0f / MZ[2 * b + 1];
    const float pb   = p[b];
    float* row = alpha + (size_t)b * SS;
    for (int s = t; s < SS; s += 256) {
        const float d = (float)s - pb;
        row[s] = __expf(row[s] - M) * invZ * __expf(-(d * d) * 0.125f);
    }
}

// ---------------------------------------------------------------------------
extern "C" void kernel_launch(void* const* d_in, const int* in_sizes, int n_in,
                              void* d_out, int out_size, void* d_ws, size_t ws_size,
                              hipStream_t stream) {
    const float* enc  = (const float*)d_in[0];   // [B,S,E]
    const float* dec  = (const float*)d_in[1];   // [B,D]
    const float* Wa_w = (const float*)d_in[2];   // [D,E]
    const float* Wa_b = (const float*)d_in[3];   // [D]
    const float* Wp_w = (const float*)d_in[4];   // [D,D]
    const float* Wp_b = (const float*)d_in[5];   // [D]
    const float* vp_w = (const float*)d_in[6];   // [D]
    const float* vp_b = (const float*)d_in[7];   // [1]

    float* out   = (float*)d_out;
    float* awe   = out;                 // [B,E]
    float* alpha = out + BB * EE;       // [B,S] (att staged here, finalized in place)

    float* ws   = (float*)d_ws;
    float* u    = ws;                   // 16384
    float* h    = ws + 16384;           // 16384
    float* p    = ws + 32768;           // 16
    float* c    = ws + 32784;           // 16
    float* marr = ws + 32800;           // 512
    float* zarr = ws + 33312;           // 512
    float* MZ   = ws + 33824;           // 32
    float* A    = ws + 33856;           // 512*1024 (2 MB), 16B aligned

    wmma_u_gemm<<<EE / 16, 32, 0, stream>>>(dec, Wa_w, u);
    wmma_h_gemm<<<DD / 16, 32, 0, stream>>>(dec, Wp_w, Wp_b, h);
    pos_c_kernel<<<1, 512, 0, stream>>>(h, vp_w, vp_b, Wa_b, dec, p, c);
    fused_att_awe<<<BB * NCH, 256, 0, stream>>>(enc, u, c, p, alpha, marr, zarr, A);
    combine_kernel<<<BB, 256, 0, stream>>>(A, marr, zarr, awe, MZ);
    alpha_kernel<<<BB, 256, 0, stream>>>(alpha, MZ, p);
}